// OptimizedGNN_77841987272808
// MI455X (gfx1250) — compile-verified
//
#include <hip/hip_runtime.h>
#include <hip/hip_bf16.h>
#include <math.h>

#define N_NODES 50000
#define N_EDGES 800000
#define D_HID   128
#define N_LAYERS 3

typedef __attribute__((ext_vector_type(2))) float v2f;
typedef __attribute__((ext_vector_type(8))) float v8f;

// ---------------------------------------------------------------------------
// GEMM: XL = X @ Wl, XR = X @ Wr   (X: [N,128], W: [128,128] row-major)
// One block = 256 threads = 8 wave32; block -> 16-row strip, wave -> 16-col tile.
// Uses V_WMMA_F32_16X16X4_F32 (fp32 in/out, matches reference numerics).
// A layout (16x4 f32): lane l holds row m=l&15, k = 2*(l>>4) + {0,1} in 2 VGPRs.
// B layout (4x16 f32): lane l holds col n=l&15, k = 2*(l>>4) + {0,1}.
// C/D layout (16x16 f32): VGPR r = row r + 8*(l>>4), col l&15.
// ---------------------------------------------------------------------------
__global__ __launch_bounds__(256) void gemm_dual_wmma(
    const float* __restrict__ X,
    const float* __restrict__ Wl,
    const float* __restrict__ Wr,
    float* __restrict__ XL,
    float* __restrict__ XR)
{
    const int wave = threadIdx.x >> 5;        // 0..7 -> col tile
    const int lane = threadIdx.x & 31;
    const int rt   = blockIdx.x;              // 0..3124 (50000/16)
    const int m    = lane & 15;
    const int koff = (lane >> 4) << 1;        // 0 or 2
    const int row  = rt * 16 + m;             // A row for this lane
    const int col  = wave * 16 + m;           // B col for this lane (n = lane&15)

    const float* __restrict__ xrow = X + (long)row * D_HID;

    v8f cl = {};
    v8f cr = {};

    #pragma unroll 4
    for (int k0 = 0; k0 < D_HID; k0 += 4) {
        const int k = k0 + koff;
        v2f a, bl, br;
        a.x  = xrow[k];
        a.y  = xrow[k + 1];
        bl.x = Wl[(long)k * D_HID + col];
        bl.y = Wl[(long)(k + 1) * D_HID + col];
        br.x = Wr[(long)k * D_HID + col];
        br.y = Wr[(long)(k + 1) * D_HID + col];
        cl = __builtin_amdgcn_wmma_f32_16x16x4_f32(false, a, false, bl,
                                                   (short)0, cl, false, false);
        cr = __builtin_amdgcn_wmma_f32_16x16x4_f32(false, a, false, br,
                                                   (short)0, cr, false, false);
    }

    const int mrow0 = rt * 16 + ((lane >> 4) << 3);   // row of VGPR 0
    const int ncol  = wave * 16 + (lane & 15);
    #pragma unroll
    for (int r = 0; r < 8; ++r) {
        XL[(long)(mrow0 + r) * D_HID + ncol] = cl[r];
        XR[(long)(mrow0 + r) * D_HID + ncol] = cr[r];
    }
}

// ---------------------------------------------------------------------------
// Per-layer scratch init: emax = -inf, denom = 0, acc = 0, stats = 0
// ---------------------------------------------------------------------------
__global__ __launch_bounds__(256) void init_layer(
    float* __restrict__ emax, float* __restrict__ denom,
    float* __restrict__ acc,  float* __restrict__ stats)
{
    const long i = (long)blockIdx.x * blockDim.x + threadIdx.x;
    if (i < (long)N_NODES * D_HID) acc[i] = 0.0f;
    if (i < N_NODES) {
        emax[i]  = -__builtin_huge_valf();
        denom[i] = 0.0f;
    }
    if (i < 2 * D_HID) stats[i] = 0.0f;
}

__device__ __forceinline__ void atomic_max_f32(float* addr, float val) {
    // ordered-int trick; emax initialized to -inf (0xFF800000)
    if (val >= 0.0f)
        atomicMax((int*)addr, __float_as_int(val));
    else
        atomicMin((unsigned int*)addr, __float_as_uint(val));
}

// ---------------------------------------------------------------------------
// Edge pass A: e = leaky_relu(xl[src] + xr[dst], 0.2) . att ;  emax[dst] = max
// One wave32 per edge; lane covers 4 channels (float4).
// ---------------------------------------------------------------------------
__global__ __launch_bounds__(256) void edge_scores(
    const float* __restrict__ XL, const float* __restrict__ XR,
    const int*  __restrict__ src, const int* __restrict__ dst,
    const float* __restrict__ att,
    float* __restrict__ e, float* __restrict__ emax)
{
    const int eid = blockIdx.x * 8 + (threadIdx.x >> 5);
    if (eid >= N_EDGES) return;
    const int lane = threadIdx.x & 31;
    const int s = src[eid];
    const int d = dst[eid];

    const float4 l  = ((const float4*)(XL + (long)s * D_HID))[lane];
    const float4 r  = ((const float4*)(XR + (long)d * D_HID))[lane];
    const float4 av = ((const float4*)att)[lane];

    float v, sum = 0.0f;
    v = l.x + r.x; v = (v > 0.0f) ? v : 0.2f * v; sum += v * av.x;
    v = l.y + r.y; v = (v > 0.0f) ? v : 0.2f * v; sum += v * av.y;
    v = l.z + r.z; v = (v > 0.0f) ? v : 0.2f * v; sum += v * av.z;
    v = l.w + r.w; v = (v > 0.0f) ? v : 0.2f * v; sum += v * av.w;

    #pragma unroll
    for (int off = 16; off > 0; off >>= 1)
        sum += __shfl_xor(sum, off, 32);

    if (lane == 0) {
        e[eid] = sum;
        atomic_max_f32(emax + d, sum);
    }
}

// ---------------------------------------------------------------------------
// Edge pass B: a = exp(e - emax[dst]); denom[dst] += a; acc[dst] += a*xl[src]
// ---------------------------------------------------------------------------
__global__ __launch_bounds__(256) void edge_aggregate(
    const float* __restrict__ XL,
    const int*  __restrict__ src, const int* __restrict__ dst,
    const float* __restrict__ e, const float* __restrict__ emax,
    float* __restrict__ denom, float* __restrict__ acc)
{
    const int eid = blockIdx.x * 8 + (threadIdx.x >> 5);
    if (eid >= N_EDGES) return;
    const int lane = threadIdx.x & 31;
    const int s = src[eid];
    const int d = dst[eid];

    const float a = expf(e[eid] - emax[d]);      // wave-uniform
    if (lane == 0) atomicAdd(denom + d, a);

    const float4 l = ((const float4*)(XL + (long)s * D_HID))[lane];
    float* pd = acc + (long)d * D_HID + lane * 4;
    atomicAdd(pd + 0, a * l.x);
    atomicAdd(pd + 1, a * l.y);
    atomicAdd(pd + 2, a * l.z);
    atomicAdd(pd + 3, a * l.w);
}

// ---------------------------------------------------------------------------
// Node pass: out = acc/(denom+1e-16) + bias (in place); column sum / sumsq
// blockDim = 128 (one thread per channel); each block handles 128 nodes.
// ---------------------------------------------------------------------------
__global__ __launch_bounds__(128) void node_out_stats(
    float* __restrict__ acc, const float* __restrict__ denom,
    const float* __restrict__ bias, float* __restrict__ stats)
{
    const int c  = threadIdx.x;
    const int n0 = blockIdx.x * 128;
    const int n1 = (n0 + 128 < N_NODES) ? (n0 + 128) : N_NODES;
    const float b = bias[c];
    float sum = 0.0f, sq = 0.0f;
    for (int n = n0; n < n1; ++n) {
        const float inv = 1.0f / (denom[n] + 1e-16f);
        const float v = acc[(long)n * D_HID + c] * inv + b;
        acc[(long)n * D_HID + c] = v;
        sum += v;
        sq  += v * v;
    }
    atomicAdd(stats + c, sum);
    atomicAdd(stats + D_HID + c, sq);
}

// ---------------------------------------------------------------------------
// BN finalize: scale = gamma*rsqrt(var+eps), shift = beta - mu*scale
// ---------------------------------------------------------------------------
__global__ __launch_bounds__(128) void bn_finalize(
    const float* __restrict__ stats,
    const float* __restrict__ gamma, const float* __restrict__ beta,
    float* __restrict__ scsh)
{
    const int c = threadIdx.x;
    const float inv_n = 1.0f / (float)N_NODES;
    const float mu  = stats[c] * inv_n;
    const float var = stats[D_HID + c] * inv_n - mu * mu;
    const float sc  = gamma[c] * rsqrtf(var + 1e-5f);
    scsh[c]         = sc;
    scsh[D_HID + c] = beta[c] - mu * sc;
}

// ---------------------------------------------------------------------------
// BN apply + exact (erf) GELU
// ---------------------------------------------------------------------------
__global__ __launch_bounds__(256) void bn_gelu(
    const float* __restrict__ out, const float* __restrict__ scsh,
    float* __restrict__ xnext)
{
    const long i = (long)blockIdx.x * blockDim.x + threadIdx.x;
    if (i >= (long)N_NODES * D_HID) return;
    const int c = (int)(i & (D_HID - 1));
    const float v = out[i] * scsh[c] + scsh[D_HID + c];
    xnext[i] = 0.5f * v * (1.0f + erff(v * 0.70710678118654752f));
}

// ---------------------------------------------------------------------------
extern "C" void kernel_launch(void* const* d_in, const int* in_sizes, int n_in,
                              void* d_out, int out_size, void* d_ws, size_t ws_size,
                              hipStream_t stream)
{
    const float* x0    = (const float*)d_in[0];   // [N,128]
    const int*   ei    = (const int*)  d_in[1];   // [2,E]
    const float* Wl    = (const float*)d_in[2];   // [3,128,128]
    const float* Wr    = (const float*)d_in[3];   // [3,128,128]
    const float* att   = (const float*)d_in[4];   // [3,128]
    const float* bias  = (const float*)d_in[5];   // [3,128]
    const float* gamma = (const float*)d_in[6];   // [3,128]
    const float* beta  = (const float*)d_in[7];   // [3,128]
    float* out = (float*)d_out;

    const long ND = (long)N_NODES * D_HID;
    float* ws    = (float*)d_ws;
    float* xl    = ws;             // N*D
    float* xr    = xl + ND;        // N*D
    float* acc   = xr + ND;        // N*D (aggregation sums, then `out`)
    float* xbuf  = acc + ND;       // N*D (inter-layer activations)
    float* e     = xbuf + ND;      // E
    float* emax  = e + N_EDGES;    // N
    float* denom = emax + N_NODES; // N
    float* stats = denom + N_NODES;// 2*128 (sum, sumsq)
    float* scsh  = stats + 2*D_HID;// 2*128 (scale, shift)

    const int* src = ei;
    const int* dst = ei + N_EDGES;

    const dim3 blk256(256), blk128(128);
    const int gemm_grid  = N_NODES / 16;                    // 3125
    const int edge_grid  = (N_EDGES + 7) / 8;               // 8 edges/block
    const int elem_grid  = (int)((ND + 255) / 256);
    const int node_grid  = (N_NODES + 127) / 128;

    const float* xin = x0;
    for (int L = 0; L < N_LAYERS; ++L) {
        const long po = (long)L * D_HID * D_HID;
        const long vo = (long)L * D_HID;
        float* xnext = (L == N_LAYERS - 1) ? out : xbuf;

        gemm_dual_wmma<<<gemm_grid, blk256, 0, stream>>>(xin, Wl + po, Wr + po, xl, xr);
        init_layer<<<elem_grid, blk256, 0, stream>>>(emax, denom, acc, stats);
        edge_scores<<<edge_grid, blk256, 0, stream>>>(xl, xr, src, dst, att + vo, e, emax);
        edge_aggregate<<<edge_grid, blk256, 0, stream>>>(xl, src, dst, e, emax, denom, acc);
        node_out_stats<<<node_grid, blk128, 0, stream>>>(acc, denom, bias + vo, stats);
        bn_finalize<<<1, blk128, 0, stream>>>(stats, gamma + vo, beta + vo, scsh);
        bn_gelu<<<elem_grid, blk256, 0, stream>>>(acc, scsh, xnext);

        xin = xnext;
    }
}